// TreeAttention_22763326668936
// MI455X (gfx1250) — compile-verified
//
#include <hip/hip_runtime.h>

#define NHEAD  8
#define TSEQ   2048
#define HID    64
#define TDENSE 512
#define AROWS  1536
#define ZC     1032
#define NEGV   (-32000.0f)

typedef __attribute__((ext_vector_type(2))) float v2f;
typedef __attribute__((ext_vector_type(8))) float v8f;

// ====================================================================
// forward_mask: one block per (head, sparse-query-row). All 6 refinement
// iterations run in LDS; emits a 2048-bit key-allow mask per row.
// ====================================================================
__global__ __launch_bounds__(256) void fmask_kernel(
    const float* __restrict__ q, const float* __restrict__ k,
    unsigned* __restrict__ bits) {
  const int r    = blockIdx.x;               // 0 .. NHEAD*AROWS-1
  const int h    = r / AROWS;
  const int a    = r - h * AROWS;
  const int tsrc = TSEQ - AROWS + 1 + a;     // 513 + a
  const int tid  = threadIdx.x;
  const float* qrow = q + ((size_t)(h * TSEQ + TDENSE + a)) * HID;
  const float* kh   = k + (size_t)h * TSEQ * HID;

  __shared__ float         qm[HID];
  __shared__ int           pix[ZC];
  __shared__ unsigned char pmk[ZC];
  __shared__ float         sc[2048];
  __shared__ int           si[2048];
  __shared__ int           ps[4096];
  __shared__ int           gtmp[ZC];
  __shared__ int           part[256];
  __shared__ unsigned      bitw[64];

  // ---- top-32 |q| mask (stable: ties keep smaller index, like lax.top_k)
  if (tid < HID) {
    float av = fabsf(qrow[tid]);
    int rank = 0;
    for (int d = 0; d < HID; ++d) {
      float ad = fabsf(qrow[d]);
      if (ad > av || (ad == av && d < tid)) ++rank;
    }
    qm[tid] = (rank < 32) ? qrow[tid] : 0.0f;
  }
  for (int z = tid; z < ZC; z += 256) {
    pix[z] = (z < 512) ? z : 0;   // where(z < k_budget, z, 0)
    pmk[z] = (z < 64) ? 1 : 0;    // z < w0
  }
  __syncthreads();

  float ws = 64.0f;
  for (int iter = 0; iter < 6; ++iter) {
    // ---- sparse-q scores against gathered k rows -----------------
    float inv = (float)tsrc / ws;
    for (int z = tid; z < 2048; z += 256) {
      float s;
      if (z < ZC) {
        if (pmk[z]) {
          int tx = (int)rintf((float)pix[z] * inv);
          tx = tx < 0 ? 0 : (tx > TSEQ - 1 ? TSEQ - 1 : tx);
          const float4* kr = (const float4*)(kh + (size_t)tx * HID);
          float acc = 0.0f;
          #pragma unroll
          for (int d4 = 0; d4 < HID / 4; ++d4) {
            float4 kv = kr[d4];
            acc += qm[4*d4+0]*kv.x + qm[4*d4+1]*kv.y
                 + qm[4*d4+2]*kv.z + qm[4*d4+3]*kv.w;
          }
          s = acc;
        } else s = NEGV;
      } else s = -3.4e38f;        // pad: sorts last
      sc[z] = s; si[z] = z;
    }
    // ---- stable descending bitonic sort of (score, idx) ----------
    for (int ksz = 2; ksz <= 2048; ksz <<= 1) {
      for (int j = ksz >> 1; j > 0; j >>= 1) {
        __syncthreads();
        for (int l = tid; l < 2048; l += 256) {
          int p = l ^ j;
          if (p > l) {
            float k0 = sc[l], k1 = sc[p];
            int   i0 = si[l], i1 = si[p];
            bool before = (k0 > k1) || (k0 == k1 && i0 < i1);
            bool up = ((l & ksz) == 0);
            if (before != up) { sc[l]=k1; sc[p]=k0; si[l]=i1; si[p]=i0; }
          }
        }
      }
    }
    __syncthreads();

    // ---- tks = clip(round(ws/tsrc*512), 1, min(ws-1, Zc-1)) ------
    float tksf = rintf(ws / (float)tsrc * 512.0f);
    float hi   = fminf(ws - 1.0f, (float)(ZC - 1));
    tksf = fminf(fmaxf(tksf, 1.0f), hi);
    int tks = (int)tksf;

    // ---- gather pixels by sorted order ---------------------------
    for (int z = tid; z < ZC; z += 256) gtmp[z] = pix[si[z]];
    __syncthreads();

    // ---- refinement: each pixel -> up to 2 children --------------
    float wsn   = fminf((float)tsrc, ws * 2.0f);
    float scale = wsn / ws;
    for (int z = tid; z < ZC; z += 256) {
      int p = gtmp[z];
      pix[z] = p;
      int m  = (z < tks) ? 1 : 0;                  // new_mask
      int p0 = (int)rintf((float)p * scale);
      int p1 = (int)rintf((float)(p + 1) * scale);
      int rep1 = p1 - p0 - 1;
      if (rep1 < 0) rep1 = 0;
      if (rep1 > m) rep1 = m;
      ps[2*z]   = p0;
      ps[2*z+1] = p0 + rep1;
    }
    for (int z = 2 * ZC + tid; z < 4096; z += 256) ps[z] = 0x7fffffff;
    // ---- ascending bitonic sort of ps ----------------------------
    for (int ksz = 2; ksz <= 4096; ksz <<= 1) {
      for (int j = ksz >> 1; j > 0; j >>= 1) {
        __syncthreads();
        for (int l = tid; l < 4096; l += 256) {
          int p = l ^ j;
          if (p > l) {
            int a0 = ps[l], b0 = ps[p];
            bool before = (a0 <= b0);
            bool up = ((l & ksz) == 0);
            if (before != up) { ps[l] = b0; ps[p] = a0; }
          }
        }
      }
    }
    __syncthreads();

    // ---- dedupe + rank scan, scatter unique values (res in si) ---
    constexpr int CHUNK = 9;            // 256*9 >= 2*ZC
    int base = tid * CHUNK;
    int loc[CHUNK];
    int run = 0;
    #pragma unroll
    for (int j = 0; j < CHUNK; ++j) {
      int i = base + j;
      int vnew = 0;
      if (i < 2 * ZC) vnew = (i == 0) || (ps[i] != ps[i - 1]);
      run += vnew; loc[j] = run;
    }
    part[tid] = run;
    for (int z = tid; z < ZC; z += 256) si[z] = -1;   // res init
    __syncthreads();
    if (tid == 0) {
      int accu = 0;
      for (int t = 0; t < 256; ++t) { int pv = part[t]; part[t] = accu; accu += pv; }
    }
    __syncthreads();
    int off = part[tid];
    #pragma unroll
    for (int j = 0; j < CHUNK; ++j) {
      int i = base + j;
      if (i < 2 * ZC) {
        int rk = off + loc[j] - 1;
        if (rk < ZC) si[rk] = ps[i];    // dup writes carry same value: benign
      }
    }
    __syncthreads();
    for (int z = tid; z < ZC; z += 256) {
      int rv = si[z];
      pmk[z] = (rv >= 0) ? 1 : 0;
      pix[z] = (rv >= 0) ? rv : 0;
    }
    ws = wsn;
    __syncthreads();
  }

  // ---- emit allow-bitmask (masked entries scatter into index 0) --
  if (tid < 64) bitw[tid] = 0;
  __syncthreads();
  for (int z = tid; z < ZC; z += 256) {
    int p = pmk[z] ? pix[z] : 0;
    if (p >= 0 && p < TSEQ) atomicOr(&bitw[p >> 5], 1u << (p & 31));
  }
  __syncthreads();
  if (tid < 64) bits[(size_t)r * 64 + tid] = bitw[tid];
}

// ====================================================================
// Flash attention with f32 WMMA (16x16x4). One wave owns 16 query rows.
// All waves in a block iterate the same number of key tiles (extra
// tiles are fully masked by the additive mask), so __syncthreads is legal.
// ====================================================================
#define NWAVES 8
#define QPB    (NWAVES * 16)

template <bool SPARSE>
__global__ __launch_bounds__(256) void attn_kernel(
    const float* __restrict__ q, const float* __restrict__ k,
    const float* __restrict__ v, const float* __restrict__ mask,
    const unsigned* __restrict__ bits, float* __restrict__ out) {
  const int chunksPerHead = (SPARSE ? AROWS : TDENSE) / QPB;
  const int h     = blockIdx.x / chunksPerHead;
  const int chunk = blockIdx.x - h * chunksPerHead;
  const int qbase = (SPARSE ? TDENSE : 0) + chunk * QPB;
  const int wv   = threadIdx.x >> 5;
  const int lane = threadIdx.x & 31;
  const int half = lane >> 4;
  const int n    = lane & 15;
  const int q0   = qbase + wv * 16;

  const float* qh = q + (size_t)h * TSEQ * HID;
  const float* kh = k + (size_t)h * TSEQ * HID;
  const float* vh = v + (size_t)h * TSEQ * HID;

  __shared__ float sP[NWAVES][16][16];
  __shared__ float rowM[NWAVES][16];
  __shared__ float rowL[NWAVES][16];
  __shared__ float rowS[NWAVES][16];

  // preload Q A-fragments: lane m=n holds K = 4c + 2*half + {0,1}
  v2f aq[16];
  {
    const float* qr = qh + (size_t)(q0 + n) * HID;
    #pragma unroll
    for (int c = 0; c < 16; ++c) {
      float2 t = *(const float2*)(qr + 4 * c + 2 * half);
      aq[c][0] = t.x; aq[c][1] = t.y;
    }
  }
  if (lane < 16) { rowM[wv][lane] = -3.0e38f; rowL[wv][lane] = 0.0f; }

  v8f acc0 = {0,0,0,0,0,0,0,0}, acc1 = {0,0,0,0,0,0,0,0};
  v8f acc2 = {0,0,0,0,0,0,0,0}, acc3 = {0,0,0,0,0,0,0,0};

  const int numKT = (qbase + QPB - 1) / 16 + 1;   // causal key-tile bound
  for (int kt = 0; kt < numKT; ++kt) {
    // ---- S = Q * K^T (16 chained f32 WMMAs over K=64) ------------
    v8f s = {0,0,0,0,0,0,0,0};
    const float* krow = kh + (size_t)(kt * 16 + n) * HID;
    #pragma unroll
    for (int c = 0; c < 16; ++c) {
      v2f b;
      b[0] = krow[4 * c + half];
      b[1] = krow[4 * c + 2 + half];
      s = __builtin_amdgcn_wmma_f32_16x16x4_f32(false, aq[c], false, b,
                                                (short)0, s, false, false);
    }
    __syncthreads();                 // prior PV reads of sP done block-wide
    const int kj = kt * 16 + n;
    #pragma unroll
    for (int j = 0; j < 8; ++j) {
      const int m  = j + 8 * half;
      const int qi = q0 + m;
      float val = s[j] + mask[(size_t)qi * TSEQ + kj];
      if (SPARSE) {
        unsigned wbit = bits[((size_t)h * AROWS + (qi - TDENSE)) * 64 + (kj >> 5)];
        val += ((wbit >> (kj & 31)) & 1u) ? 0.0f : NEGV;
      }
      sP[wv][m][n] = val;
    }
    __syncthreads();
    // ---- online softmax: lanes 0..15 own one row each ------------
    if (lane < 16) {
      float mOld = rowM[wv][lane];
      float mNew = mOld;
      #pragma unroll
      for (int t = 0; t < 16; ++t) mNew = fmaxf(mNew, sP[wv][lane][t]);
      float corr = expf(mOld - mNew);
      float lsum = rowL[wv][lane] * corr;
      #pragma unroll
      for (int t = 0; t < 16; ++t) {
        float pe = expf(sP[wv][lane][t] - mNew);
        sP[wv][lane][t] = pe;
        lsum += pe;
      }
      rowM[wv][lane] = mNew; rowL[wv][lane] = lsum; rowS[wv][lane] = corr;
    }
    __syncthreads();
    // ---- rescale accumulators, then O += P * V (f32 WMMA) --------
    #pragma unroll
    for (int j = 0; j < 8; ++j) {
      float f = rowS[wv][j + 8 * half];
      acc0[j] *= f; acc1[j] *= f; acc2[j] *= f; acc3[j] *= f;
    }
    #pragma unroll
    for (int kc = 0; kc < 4; ++kc) {
      v2f ap;
      ap[0] = sP[wv][n][4 * kc + 2 * half];
      ap[1] = sP[wv][n][4 * kc + 2 * half + 1];
      const float* v0 = vh + (size_t)(kt * 16 + 4 * kc + half) * HID;
      const float* v1 = vh + (size_t)(kt * 16 + 4 * kc + 2 + half) * HID;
      v2f b;
      b[0] = v0[n];      b[1] = v1[n];
      acc0 = __builtin_amdgcn_wmma_f32_16x16x4_f32(false, ap, false, b, (short)0, acc0, false, false);
      b[0] = v0[16 + n]; b[1] = v1[16 + n];
      acc1 = __builtin_amdgcn_wmma_f32_16x16x4_f32(false, ap, false, b, (short)0, acc1, false, false);
      b[0] = v0[32 + n]; b[1] = v1[32 + n];
      acc2 = __builtin_amdgcn_wmma_f32_16x16x4_f32(false, ap, false, b, (short)0, acc2, false, false);
      b[0] = v0[48 + n]; b[1] = v1[48 + n];
      acc3 = __builtin_amdgcn_wmma_f32_16x16x4_f32(false, ap, false, b, (short)0, acc3, false, false);
    }
  }
  // ---- normalize and store ---------------------------------------
  #pragma unroll
  for (int j = 0; j < 8; ++j) {
    const int m = j + 8 * half;
    const float invl = 1.0f / rowL[wv][m];
    float* op = out + ((size_t)h * TSEQ + (q0 + m)) * HID + n;
    op[0]  = acc0[j] * invl;
    op[16] = acc1[j] * invl;
    op[32] = acc2[j] * invl;
    op[48] = acc3[j] * invl;
  }
}

// ====================================================================
extern "C" void kernel_launch(void* const* d_in, const int* in_sizes, int n_in,
                              void* d_out, int out_size, void* d_ws, size_t ws_size,
                              hipStream_t stream) {
  (void)in_sizes; (void)n_in; (void)out_size; (void)ws_size;
  const float* q    = (const float*)d_in[0];
  const float* k    = (const float*)d_in[1];
  const float* v    = (const float*)d_in[2];
  const float* mask = (const float*)d_in[3];
  float*    out  = (float*)d_out;
  unsigned* bits = (unsigned*)d_ws;   // NHEAD*AROWS*64 u32 = 3 MB scratch

  // dense branch (independent of mask bits)
  attn_kernel<false><<<NHEAD * (TDENSE / QPB), 256, 0, stream>>>(q, k, v, mask, nullptr, out);
  // forward_mask -> per-row key bitmask
  fmask_kernel<<<NHEAD * AROWS, 256, 0, stream>>>(q, k, bits);
  // sparse branch gated by bitmask
  attn_kernel<true><<<NHEAD * (AROWS / QPB), 256, 0, stream>>>(q, k, v, mask, bits, out);
}